// AttentionCopy_21552145891837
// MI455X (gfx1250) — compile-verified
//
#include <hip/hip_runtime.h>
#include <hip/hip_bf16.h>

#define VOCAB   50000
#define IN_LEN  512
#define OUT_LEN 32
#define BATCH   16

// ---------------------------------------------------------------------------
// Kernel 1: zero the 102.4 MB output with 128-bit stores (global_store_b128).
// Block-contiguous tiling: each block owns 256*8 float4 = 32 KB; each thread
// issues 8 b128 stores at immediate offsets (j*256 float4 = 4096 B, fits the
// signed 24-bit ioffset) -> near-zero loop overhead, pure store bandwidth.
// ---------------------------------------------------------------------------
constexpr int ZBLOCK  = 256;
constexpr int ZUNROLL = 8;

__global__ void zero_fill_kernel(float* __restrict__ out, long long n4, long long ntail)
{
    const long long chunk = (long long)ZBLOCK * ZUNROLL;
    const long long base  = (long long)blockIdx.x * chunk + threadIdx.x;
    float4* __restrict__ out4 = (float4*)out;
    const float4 z = make_float4(0.f, 0.f, 0.f, 0.f);

    if (base + (long long)(ZUNROLL - 1) * ZBLOCK < n4) {
        // fast path: full tile, 8 stores with immediate offsets
#pragma unroll
        for (int j = 0; j < ZUNROLL; ++j)
            out4[base + (long long)j * ZBLOCK] = z;
    } else {
        // edge tile
#pragma unroll
        for (int j = 0; j < ZUNROLL; ++j) {
            const long long idx = base + (long long)j * ZBLOCK;
            if (idx < n4) out4[idx] = z;
        }
    }
    // tail (out_size % 4) — zero for this problem, kept generic
    if (blockIdx.x == 0 && (long long)threadIdx.x < ntail)
        out[n4 * 4 + threadIdx.x] = 0.f;
}

// ---------------------------------------------------------------------------
// Kernel 2: weighted scatter-add.
// One thread per (batch, i) pair; lanes run along i so ids[] and each fixed-o
// row of attn[] are coalesced. Phase 1 batches all 32 attn loads (one wait,
// full MLP), phase 2 fires 32 hardware global_atomic_add_f32 back-to-back.
// Duplicate ids require atomic accumulation; 262144 4B atomics resolve in L2
// (the whole 102.4 MB output fits in the 192 MB L2).
// ---------------------------------------------------------------------------
__global__ void scatter_add_kernel(const int*   __restrict__ ids,
                                   const float* __restrict__ attn,
                                   float*       __restrict__ out)
{
    const int t = blockIdx.x * blockDim.x + threadIdx.x;   // 0 .. 8191
    const int b = t >> 9;          // t / 512
    const int i = t & (IN_LEN - 1);

    const int id = ids[t];         // coalesced across lanes

    const float* __restrict__ ab = attn + (size_t)b * OUT_LEN * IN_LEN + i;
    float*       __restrict__ ob = out  + (size_t)b * OUT_LEN * VOCAB  + id;

    // phase 1: gather the column attn[b][:, i] into registers (batched loads)
    float vals[OUT_LEN];
#pragma unroll
    for (int o = 0; o < OUT_LEN; ++o)
        vals[o] = ab[(size_t)o * IN_LEN];      // imm offset o*2048 B

    // phase 2: scatter with native fp32 atomics (imm offset o*200000 B)
#pragma unroll
    for (int o = 0; o < OUT_LEN; ++o)
        unsafeAtomicAdd(ob + (size_t)o * VOCAB, vals[o]);
}

extern "C" void kernel_launch(void* const* d_in, const int* in_sizes, int n_in,
                              void* d_out, int out_size, void* d_ws, size_t ws_size,
                              hipStream_t stream)
{
    const int*   ids  = (const int*)  d_in[0];   // [16, 512] int32
    const float* attn = (const float*)d_in[1];   // [16, 32, 512] f32
    float*       out  = (float*)      d_out;     // [16, 32, 50000] f32

    // --- pass 1: zero the output ---
    const long long n4    = (long long)out_size >> 2;   // 6,400,000 float4
    const long long ntail = (long long)out_size & 3LL;
    const long long chunk = (long long)ZBLOCK * ZUNROLL;
    long long zblocks = (n4 + chunk - 1) / chunk;       // 3125 blocks
    if (zblocks < 1) zblocks = 1;
    zero_fill_kernel<<<(int)zblocks, ZBLOCK, 0, stream>>>(out, n4, ntail);

    // --- pass 2: scatter-add (stream-ordered after the fill) ---
    const int total = BATCH * IN_LEN;                   // 8192 threads
    scatter_add_kernel<<<total / 256, 256, 0, stream>>>(ids, attn, out);
}